// GCN_71236327571563
// MI455X (gfx1250) — compile-verified
//
#include <hip/hip_runtime.h>

typedef __attribute__((ext_vector_type(2))) float v2f;
typedef __attribute__((ext_vector_type(8))) float v8f;

#define IN_DIM  128
#define HID_DIM 128
#define OUT_DIM 64

// ---------------------------------------------------------------- utilities

__global__ __launch_bounds__(256) void zero_f32(float* __restrict__ p, long n) {
  long i = (long)blockIdx.x * blockDim.x + threadIdx.x;
  if (i < n) p[i] = 0.0f;
}

__global__ __launch_bounds__(256) void degree_kernel(const int* __restrict__ src,
                                                     const int* __restrict__ dst,
                                                     float* __restrict__ out_deg,
                                                     float* __restrict__ in_deg,
                                                     int n_edges) {
  int i = blockIdx.x * blockDim.x + threadIdx.x;
  if (i < n_edges) {
    atomicAdd(&out_deg[src[i]], 1.0f);
    atomicAdd(&in_deg[dst[i]], 1.0f);
  }
}

__global__ __launch_bounds__(256) void norm_kernel(const float* __restrict__ out_deg,
                                                   const float* __restrict__ in_deg,
                                                   float* __restrict__ out_norm,
                                                   float* __restrict__ in_norm,
                                                   int n) {
  int i = blockIdx.x * blockDim.x + threadIdx.x;
  if (i < n) {
    out_norm[i] = rsqrtf(fmaxf(out_deg[i], 1.0f));
    in_norm[i]  = rsqrtf(fmaxf(in_deg[i], 1.0f));
  }
}

// ------------------------------------------------------------- WMMA GEMM 1
// h1[m, n] = sum_k (x[m,k] * out_norm[m]) * W1[k, n]
// One wave computes one 16x16 tile with V_WMMA_F32_16X16X4_F32 (fp32-exact).
// A fragment (16x4 f32): lanes 0-15 hold K = kbase+0 (v0), kbase+1 (v1);
// lanes 16-31 hold K = kbase+2 (v0), kbase+3 (v1). B (4x16) mirrored.
// C/D: VGPR r holds row M = 16*rt + 8*(lane/16) + r, col N = lane%16.

__global__ __launch_bounds__(256) void gemm1_wmma(const float* __restrict__ x,
                                                  const float* __restrict__ W1,
                                                  const float* __restrict__ out_norm,
                                                  float* __restrict__ h1,
                                                  int n_row_tiles) {
  const int lane = threadIdx.x & 31;
  const int wave = (int)((blockIdx.x * blockDim.x + threadIdx.x) >> 5);
  const int n_tiles = n_row_tiles * (HID_DIM / 16);
  if (wave >= n_tiles) return;                 // wave-uniform: EXEC stays all-1s
  const int rt   = wave / (HID_DIM / 16);
  const int ct   = wave % (HID_DIM / 16);
  const int half = lane >> 4;                  // 0 or 1
  const int sub  = lane & 15;
  const int m    = rt * 16 + sub;              // A row for this lane
  const int n    = ct * 16 + sub;              // B/D column for this lane
  const int kh   = half * 2;                   // K sub-offset 0 or 2

  const float norm = out_norm[m];
  const float* __restrict__ arow = x + (size_t)m * IN_DIM;

  v8f acc = {};
#pragma unroll 4
  for (int k = 0; k < IN_DIM; k += 4) {
    const float2 av = *(const float2*)(arow + k + kh);
    v2f a, b;
    a.x = av.x * norm;
    a.y = av.y * norm;
    b.x = W1[(size_t)(k + kh)     * HID_DIM + n];
    b.y = W1[(size_t)(k + kh + 1) * HID_DIM + n];
    acc = __builtin_amdgcn_wmma_f32_16x16x4_f32(
        /*neg_a=*/false, a, /*neg_b=*/false, b,
        /*c_mod=*/(short)0, acc, /*reuse_a=*/false, /*reuse_b=*/false);
  }

  const int m_base = rt * 16 + half * 8;
#pragma unroll
  for (int r = 0; r < 8; ++r)
    h1[(size_t)(m_base + r) * HID_DIM + n] = acc[r];
}

// ------------------------------------------------------------- WMMA GEMM 2
// A[m,k] = relu(agg1[m,k] * in_norm[m] + b1[k]) * out_norm[m];  D = A @ W2

__global__ __launch_bounds__(256) void gemm2_wmma(const float* __restrict__ agg1,
                                                  const float* __restrict__ W2,
                                                  const float* __restrict__ b1,
                                                  const float* __restrict__ out_norm,
                                                  const float* __restrict__ in_norm,
                                                  float* __restrict__ h2,
                                                  int n_row_tiles) {
  const int lane = threadIdx.x & 31;
  const int wave = (int)((blockIdx.x * blockDim.x + threadIdx.x) >> 5);
  const int n_tiles = n_row_tiles * (OUT_DIM / 16);
  if (wave >= n_tiles) return;
  const int rt   = wave / (OUT_DIM / 16);
  const int ct   = wave % (OUT_DIM / 16);
  const int half = lane >> 4;
  const int sub  = lane & 15;
  const int m    = rt * 16 + sub;
  const int n    = ct * 16 + sub;
  const int kh   = half * 2;

  const float s_in  = in_norm[m];
  const float s_out = out_norm[m];
  const float* __restrict__ arow = agg1 + (size_t)m * HID_DIM;

  v8f acc = {};
#pragma unroll 4
  for (int k = 0; k < HID_DIM; k += 4) {
    const float2 av = *(const float2*)(arow + k + kh);
    const float2 bv = *(const float2*)(b1 + k + kh);
    v2f a, b;
    a.x = fmaxf(av.x * s_in + bv.x, 0.0f) * s_out;
    a.y = fmaxf(av.y * s_in + bv.y, 0.0f) * s_out;
    b.x = W2[(size_t)(k + kh)     * OUT_DIM + n];
    b.y = W2[(size_t)(k + kh + 1) * OUT_DIM + n];
    acc = __builtin_amdgcn_wmma_f32_16x16x4_f32(
        false, a, false, b, (short)0, acc, false, false);
  }

  const int m_base = rt * 16 + half * 8;
#pragma unroll
  for (int r = 0; r < 8; ++r)
    h2[(size_t)(m_base + r) * OUT_DIM + n] = acc[r];
}

// ------------------------------------------------------------ edge scatter
// One wave per edge. 128-dim: each lane moves 4 consecutive floats (float4
// gather + 4 global_atomic_add_f32). agg fits in L2 (25.6 MB), so the atomic
// RMW stays on-die.

__global__ __launch_bounds__(256) void scatter128(const float* __restrict__ h,
                                                  const int* __restrict__ src,
                                                  const int* __restrict__ dst,
                                                  float* __restrict__ agg,
                                                  int n_edges) {
  const int wave = (int)((blockIdx.x * blockDim.x + threadIdx.x) >> 5);
  const int lane = threadIdx.x & 31;
  if (wave >= n_edges) return;
  const int s = src[wave];
  const int d = dst[wave];
  const float4 v = *(const float4*)(h + (size_t)s * 128 + lane * 4);
  float* p = agg + (size_t)d * 128 + lane * 4;
  atomicAdd(p + 0, v.x);
  atomicAdd(p + 1, v.y);
  atomicAdd(p + 2, v.z);
  atomicAdd(p + 3, v.w);
}

// 64-dim: each lane moves 2 consecutive floats.
__global__ __launch_bounds__(256) void scatter64(const float* __restrict__ h,
                                                 const int* __restrict__ src,
                                                 const int* __restrict__ dst,
                                                 float* __restrict__ agg,
                                                 int n_edges) {
  const int wave = (int)((blockIdx.x * blockDim.x + threadIdx.x) >> 5);
  const int lane = threadIdx.x & 31;
  if (wave >= n_edges) return;
  const int s = src[wave];
  const int d = dst[wave];
  const float2 v = *(const float2*)(h + (size_t)s * 64 + lane * 2);
  float* p = agg + (size_t)d * 64 + lane * 2;
  atomicAdd(p + 0, v.x);
  atomicAdd(p + 1, v.y);
}

// out[m,n] = out[m,n] * in_norm[m] + b2[n]   (in place on d_out)
__global__ __launch_bounds__(256) void finalize_kernel(float* __restrict__ out,
                                                       const float* __restrict__ in_norm,
                                                       const float* __restrict__ b2,
                                                       long total) {
  long i = (long)blockIdx.x * blockDim.x + threadIdx.x;
  if (i < total) {
    const int m = (int)(i >> 6);
    const int n = (int)(i & 63);
    out[i] = out[i] * in_norm[m] + b2[n];
  }
}

// ----------------------------------------------------------------- launcher

static inline int grid1d(long n, int block) { return (int)((n + block - 1) / block); }

extern "C" void kernel_launch(void* const* d_in, const int* in_sizes, int n_in,
                              void* d_out, int out_size, void* d_ws, size_t ws_size,
                              hipStream_t stream) {
  const float* x   = (const float*)d_in[0];
  const float* W1  = (const float*)d_in[1];
  const float* b1  = (const float*)d_in[2];
  const float* W2  = (const float*)d_in[3];
  const float* b2  = (const float*)d_in[4];
  const int*   src = (const int*)d_in[5];
  const int*   dst = (const int*)d_in[6];

  const int n_nodes = in_sizes[0] / IN_DIM;   // 50000
  const int n_edges = in_sizes[5];            // 1.6M
  float* out = (float*)d_out;

  // workspace layout (floats): deg(2N) norms(2N) h1(128N) agg1(128N) h2(64N)
  float* ws       = (float*)d_ws;
  float* out_deg  = ws;
  float* in_deg   = out_deg + (size_t)n_nodes;
  float* out_norm = in_deg + (size_t)n_nodes;
  float* in_norm  = out_norm + (size_t)n_nodes;
  float* h1       = in_norm + (size_t)n_nodes;
  float* agg1     = h1 + (size_t)n_nodes * HID_DIM;
  float* h2       = agg1 + (size_t)n_nodes * HID_DIM;

  // zero accumulators (harness does not re-poison between replays)
  zero_f32<<<grid1d(2L * n_nodes, 256), 256, 0, stream>>>(out_deg, 2L * n_nodes);
  zero_f32<<<grid1d((long)n_nodes * HID_DIM, 256), 256, 0, stream>>>(agg1, (long)n_nodes * HID_DIM);
  zero_f32<<<grid1d((long)n_nodes * OUT_DIM, 256), 256, 0, stream>>>(out, (long)n_nodes * OUT_DIM);

  degree_kernel<<<grid1d(n_edges, 256), 256, 0, stream>>>(src, dst, out_deg, in_deg, n_edges);
  norm_kernel<<<grid1d(n_nodes, 256), 256, 0, stream>>>(out_deg, in_deg, out_norm, in_norm, n_nodes);

  const int row_tiles = n_nodes / 16;  // 50000 % 16 == 0

  {
    const int waves  = row_tiles * (HID_DIM / 16);
    const int blocks = (waves + 7) / 8;  // 8 waves per 256-thread block
    gemm1_wmma<<<blocks, 256, 0, stream>>>(x, W1, out_norm, h1, row_tiles);
  }

  scatter128<<<(n_edges + 7) / 8, 256, 0, stream>>>(h1, src, dst, agg1, n_edges);

  {
    const int waves  = row_tiles * (OUT_DIM / 16);
    const int blocks = (waves + 7) / 8;
    gemm2_wmma<<<blocks, 256, 0, stream>>>(agg1, W2, b1, out_norm, in_norm, h2, row_tiles);
  }

  scatter64<<<(n_edges + 7) / 8, 256, 0, stream>>>(h2, src, dst, out, n_edges);

  finalize_kernel<<<grid1d((long)n_nodes * OUT_DIM, 256), 256, 0, stream>>>(
      out, in_norm, b2, (long)n_nodes * OUT_DIM);
}